// PatientRepresentationGNN_15281493639280
// MI455X (gfx1250) — compile-verified
//
#include <hip/hip_runtime.h>
#include <hip/hip_bf16.h>

#define N_PAT 50000
#define N_RAD 100000
#define N_GEN 20000
#define NEDGE 800000
#define HDIM  64
#define OUTD  32
#define NEG_SLOPE 0.2f

typedef __attribute__((ext_vector_type(2))) float v2f;
typedef __attribute__((ext_vector_type(8))) float v8f;

// ---------------------------------------------------------------------------
// WMMA f32 GEMM: C = op(A) @ B (+bias). A: MxK row-major, B: KxN row-major.
// N = NT*16 is a compile-time constant; one wave computes a full 16xN row
// block: the A fragment is loaded once per K-step and reused for NT
// V_WMMA_F32_16X16X4_F32 issues. Requires M%16==0, K%4==0 (true here).
// ---------------------------------------------------------------------------
template <int NT, bool RELU_A>
__global__ void gemm_wmma_f32(const float* __restrict__ A,
                              const float* __restrict__ B,
                              float* __restrict__ C,
                              int M, int K,
                              const float* __restrict__ bias)
{
    constexpr int N = NT * 16;
    const int lane = threadIdx.x & 31;
    const int wid  = blockIdx.x * (blockDim.x >> 5) + (threadIdx.x >> 5);
    if (wid >= (M >> 4)) return;                 // wave-uniform: EXEC stays all-1s
    const int m0 = wid << 4;
    const int g  = lane >> 4;                    // lane group (0/1)
    const int l  = lane & 15;

    v8f acc[NT] = {};
    const float* arow = A + (size_t)(m0 + l) * K + (g << 1);

    for (int k = 0; k < K; k += 4) {
        // A fragment: component r holds element (m0+l, k + 2*g + r)
        v2f a;
        a.x = arow[k];
        a.y = arow[k + 1];
        if (RELU_A) { a.x = fmaxf(a.x, 0.f); a.y = fmaxf(a.y, 0.f); }
        // B fragments: component r holds element (k + 2*g + r, 16*t + l)
        const float* brow = B + (size_t)(k + (g << 1)) * N + l;
#pragma unroll
        for (int t = 0; t < NT; ++t) {
            v2f b;
            b.x = brow[t * 16];
            b.y = brow[t * 16 + N];
            acc[t] = __builtin_amdgcn_wmma_f32_16x16x4_f32(
                false, a, false, b, (short)0, acc[t], false, false);
        }
    }

#pragma unroll
    for (int t = 0; t < NT; ++t) {
        const float bv = bias ? bias[t * 16 + l] : 0.f;
        // D fragment: component r holds element (m0 + r + 8*g, 16*t + l)
#pragma unroll
        for (int r = 0; r < 8; ++r)
            C[(size_t)(m0 + r + (g << 3)) * N + t * 16 + l] = acc[t][r] + bv;
    }
}

// ---------------------------------------------------------------------------
// wv[k] = sum_h W[k,h] * a[h]   (fold attention vector through weight matrix)
// ---------------------------------------------------------------------------
__global__ void fold_w(const float* __restrict__ W, const float* __restrict__ a,
                       float* __restrict__ wv, int K)
{
    int k = blockIdx.x * blockDim.x + threadIdx.x;
    if (k >= K) return;
    float s = 0.f;
    for (int h = 0; h < HDIM; ++h) s += W[(size_t)k * HDIM + h] * a[h];
    wv[k] = s;
}

// ---------------------------------------------------------------------------
// y[row] = dot(X[row,:], w)  — one wave per row, shuffle reduction (wave32)
// ---------------------------------------------------------------------------
__global__ void gemv_rows(const float* __restrict__ X, const float* __restrict__ w,
                          float* __restrict__ y, int Nrows, int K)
{
    int row = blockIdx.x * (blockDim.x >> 5) + (threadIdx.x >> 5);
    if (row >= Nrows) return;
    int lane = threadIdx.x & 31;
    float s = 0.f;
    for (int k = lane; k < K; k += 32) s += X[(size_t)row * K + k] * w[k];
#pragma unroll
    for (int off = 16; off; off >>= 1) s += __shfl_xor(s, off);
    if (lane == 0) y[row] = s;
}

// ---- monotone float <-> uint encoding for atomic float-max ----------------
__device__ __forceinline__ unsigned f2ord(float f) {
    unsigned u = __float_as_uint(f);
    return (u & 0x80000000u) ? ~u : (u | 0x80000000u);
}
__device__ __forceinline__ float ord2f(unsigned u) {
    return __uint_as_float((u & 0x80000000u) ? (u ^ 0x80000000u) : ~u);
}

__global__ void init_acc(const float* __restrict__ b0, const float* __restrict__ b1,
                         const float* __restrict__ b2, float* __restrict__ acc)
{
    int i = blockIdx.x * blockDim.x + threadIdx.x;
    if (i >= N_PAT * HDIM) return;
    int h = i & (HDIM - 1);
    acc[i] = b0[h] + b1[h] + b2[h];
}

__global__ void zero_seg(unsigned* __restrict__ m_ord, float* __restrict__ denom)
{
    int i = blockIdx.x * blockDim.x + threadIdx.x;
    if (i >= N_PAT) return;
    m_ord[i] = 0u;      // decodes to NaN => treated as empty segment => m=0
    denom[i] = 0.f;
}

// e = leaky_relu(alpha_s[src] + alpha_d[dst]); segment max via ordered-uint max
__global__ void edge_pass1(const int* __restrict__ src, const int* __restrict__ dst,
                           const float* __restrict__ als, const float* __restrict__ ald,
                           float* __restrict__ ebuf, unsigned* __restrict__ m_ord)
{
    int i = blockIdx.x * blockDim.x + threadIdx.x;
    if (i >= NEDGE) return;
    float e = als[src[i]] + ald[dst[i]];
    e = (e > 0.f) ? e : NEG_SLOPE * e;
    ebuf[i] = e;
    atomicMax(&m_ord[dst[i]], f2ord(e));
}

// ex = exp(e - m[dst]); segment sum of ex
__global__ void edge_pass2(const int* __restrict__ dst,
                           const unsigned* __restrict__ m_ord,
                           float* __restrict__ ebuf, float* __restrict__ denom)
{
    int i = blockIdx.x * blockDim.x + threadIdx.x;
    if (i >= NEDGE) return;
    int d = dst[i];
    float m = ord2f(m_ord[d]);
    if (!(m >= -3.0e38f && m <= 3.0e38f)) m = 0.f;   // NaN / inf -> 0 (empty seg)
    float ex = expf(ebuf[i] - m);
    ebuf[i] = ex;
    atomicAdd(&denom[d], ex);
}

// acc[dst,h] += hs[src,h] * ex / max(denom[dst],1e-16) — thread per (edge,h)
__global__ void edge_pass3(const int* __restrict__ src, const int* __restrict__ dst,
                           const float* __restrict__ ebuf, const float* __restrict__ denom,
                           const float* __restrict__ hs, float* __restrict__ acc)
{
    long long t = (long long)blockIdx.x * blockDim.x + threadIdx.x;
    if (t >= (long long)NEDGE * HDIM) return;
    int i = (int)(t >> 6);
    int h = (int)(t & (HDIM - 1));
    int d = dst[i];
    float alpha = ebuf[i] / fmaxf(denom[d], 1e-16f);
    atomicAdd(&acc[(size_t)d * HDIM + h], hs[(size_t)src[i] * HDIM + h] * alpha);
}

// ---------------------------------------------------------------------------
extern "C" void kernel_launch(void* const* d_in, const int* in_sizes, int n_in,
                              void* d_out, int out_size, void* d_ws, size_t ws_size,
                              hipStream_t stream)
{
    const float* x_pat = (const float*)d_in[0];
    const float* x_rad = (const float*)d_in[1];
    const float* x_gen = (const float*)d_in[2];
    const int* src_rp = (const int*)d_in[3];
    const int* dst_rp = (const int*)d_in[4];
    const int* src_gp = (const int*)d_in[5];
    const int* dst_gp = (const int*)d_in[6];
    const int* src_pp = (const int*)d_in[7];
    const int* dst_pp = (const int*)d_in[8];
    const float* Ws_rp = (const float*)d_in[9];
    const float* Wd_rp = (const float*)d_in[10];
    const float* as_rp = (const float*)d_in[11];
    const float* ad_rp = (const float*)d_in[12];
    const float* b_rp  = (const float*)d_in[13];
    const float* Ws_gp = (const float*)d_in[14];
    const float* Wd_gp = (const float*)d_in[15];
    const float* as_gp = (const float*)d_in[16];
    const float* ad_gp = (const float*)d_in[17];
    const float* b_gp  = (const float*)d_in[18];
    const float* Ws_pp = (const float*)d_in[19];
    const float* Wd_pp = (const float*)d_in[20];
    const float* as_pp = (const float*)d_in[21];
    const float* ad_pp = (const float*)d_in[22];
    const float* b_pp  = (const float*)d_in[23];
    const float* W_lin = (const float*)d_in[24];
    const float* b_lin = (const float*)d_in[25];
    float* out = (float*)d_out;

    // ---- workspace carve-up (~43 MB) ----
    char* ws = (char*)d_ws;
    size_t off = 0;
    auto carve = [&](size_t bytes) -> void* {
        void* p = ws + off;
        off += (bytes + 255) & ~(size_t)255;
        return p;
    };
    float*    hs    = (float*)carve((size_t)N_RAD * HDIM * sizeof(float)); // max Ns
    float*    acc   = (float*)carve((size_t)N_PAT * HDIM * sizeof(float));
    float*    ebuf  = (float*)carve((size_t)NEDGE * sizeof(float));
    float*    alps  = (float*)carve((size_t)N_RAD * sizeof(float));
    float*    alpd  = (float*)carve((size_t)N_PAT * sizeof(float));
    unsigned* mord  = (unsigned*)carve((size_t)N_PAT * sizeof(unsigned));
    float*    denom = (float*)carve((size_t)N_PAT * sizeof(float));
    float*    wsv   = (float*)carve(512 * sizeof(float));
    float*    wdv   = (float*)carve(512 * sizeof(float));
    (void)ws_size; (void)in_sizes; (void)n_in; (void)out_size;

    const dim3 blk(256);

    init_acc<<<(N_PAT * HDIM + 255) / 256, blk, 0, stream>>>(b_rp, b_gp, b_pp, acc);

    auto run_layer = [&](const float* xs, int Ns, int Ks,
                         const int* src, const int* dst,
                         const float* Ws, const float* Wd,
                         const float* a_s, const float* a_d) {
        int waves = Ns / 16;   // one wave per 16xH row block (NT=4 tiles)
        gemm_wmma_f32<HDIM / 16, false>
            <<<(waves + 7) / 8, blk, 0, stream>>>(xs, Ws, hs, Ns, Ks, nullptr);
        fold_w<<<(Ks + 127) / 128, dim3(128), 0, stream>>>(Ws, a_s, wsv, Ks);
        fold_w<<<1, dim3(128), 0, stream>>>(Wd, a_d, wdv, 128);
        gemv_rows<<<(Ns + 7) / 8, blk, 0, stream>>>(xs, wsv, alps, Ns, Ks);
        gemv_rows<<<(N_PAT + 7) / 8, blk, 0, stream>>>(x_pat, wdv, alpd, N_PAT, 128);
        zero_seg<<<(N_PAT + 255) / 256, blk, 0, stream>>>(mord, denom);
        edge_pass1<<<(NEDGE + 255) / 256, blk, 0, stream>>>(src, dst, alps, alpd, ebuf, mord);
        edge_pass2<<<(NEDGE + 255) / 256, blk, 0, stream>>>(dst, mord, ebuf, denom);
        long long t3 = (long long)NEDGE * HDIM;
        edge_pass3<<<(unsigned)((t3 + 255) / 256), blk, 0, stream>>>(src, dst, ebuf, denom, hs, acc);
    };

    run_layer(x_rad, N_RAD, 256, src_rp, dst_rp, Ws_rp, Wd_rp, as_rp, ad_rp);
    run_layer(x_gen, N_GEN, 512, src_gp, dst_gp, Ws_gp, Wd_gp, as_gp, ad_gp);
    run_layer(x_pat, N_PAT, 128, src_pp, dst_pp, Ws_pp, Wd_pp, as_pp, ad_pp);

    // out = relu(acc) @ W_lin + b_lin   (M=50000, N=32 -> NT=2, K=64)
    int waves = N_PAT / 16;
    gemm_wmma_f32<OUTD / 16, true>
        <<<(waves + 7) / 8, blk, 0, stream>>>(acc, W_lin, out, N_PAT, HDIM, b_lin);
}